// SideGating_59803124629520
// MI455X (gfx1250) — compile-verified
//
#include <hip/hip_runtime.h>

typedef __attribute__((ext_vector_type(2))) float v2f;
typedef __attribute__((ext_vector_type(8))) float v8f;

#define W_MIN 0.005f
#define W_MAX 0.02f
#define LOG2E 1.4426950408889634f

// One wave32 computes one 16x16 output tile via three V_WMMA_F32_16X16X4_F32
// (K=3 padded to 4) plus a fused clip/sqrt/sigmoid epilogue.
// - Operand loads are unconditional (v_cndmask component select, no EXEC
//   divergence), B-column point loads reused for |p_j|^2.
// - clamp is hoisted BEFORE the sqrt (monotone), so the raw single-instruction
//   v_sqrt_f32 suffices (no IEEE denorm-fixup sequence) and the >=0 guard is
//   subsumed by the lower clamp bound.
__global__ __launch_bounds__(256) void SideGating_wmma_kernel(
    const float* __restrict__ points,    // [B,N,3]
    const float* __restrict__ pdir,      // [B,N,3]
    const float* __restrict__ normals,   // [B,N,3]
    const float* __restrict__ density,   // [B,N]
    const float* __restrict__ a1p,
    const float* __restrict__ a2p,
    const float* __restrict__ a3p,
    float* __restrict__ out,             // [B,N,N]
    int N, int tilesPerSide)
{
    const int lane = threadIdx.x & 31;
    const int wave = threadIdx.x >> 5;
    const int tile = blockIdx.x * (blockDim.x >> 5) + wave;

    const int tilesPerBatch = tilesPerSide * tilesPerSide;
    const int b  = tile / tilesPerBatch;
    const int t  = tile - b * tilesPerBatch;
    const int ti = t / tilesPerSide;
    const int tj = t - ti * tilesPerSide;

    const int i0 = ti << 4;
    const int j0 = tj << 4;

    const float* __restrict__ Pb = points  + (size_t)b * N * 3;
    const float* __restrict__ Db = pdir    + (size_t)b * N * 3;
    const float* __restrict__ Nb = normals + (size_t)b * N * 3;
    const float* __restrict__ Rb = density + (size_t)b * N;

    const int half = lane >> 4;   // 0: K=(0,1)   1: K=(2,3=pad)
    const int l16  = lane & 15;

    // ---- unconditional component loads for A row (i0+l16) ----
    const int arow = (i0 + l16) * 3;
    const float apx = Pb[arow + 0], apy = Pb[arow + 1], apz = Pb[arow + 2];
    const float adx = Db[arow + 0], ady = Db[arow + 1], adz = Db[arow + 2];
    const float anx = Nb[arow + 0], any_ = Nb[arow + 1], anz = Nb[arow + 2];

    // ---- unconditional component loads for B col (j0+l16) ----
    const int j    = j0 + l16;
    const int brow = j * 3;
    const float bpx = Pb[brow + 0], bpy = Pb[brow + 1], bpz = Pb[brow + 2];
    const float bnx = Nb[brow + 0], bny = Nb[brow + 1], bnz = Nb[brow + 2];

    // A layout (32-bit A, 16x4): lanes 0-15 hold K=0,1 ; lanes 16-31 hold K=2,3(pad)
    v2f aP, aD, aN, bP, bN;
    aP.x = half ? apz : apx;   aP.y = half ? 0.0f : apy;
    aD.x = half ? adz : adx;   aD.y = half ? 0.0f : ady;
    aN.x = half ? anz : anx;   aN.y = half ? 0.0f : any_;
    bP.x = half ? bpz : bpx;   bP.y = half ? 0.0f : bpy;
    bN.x = half ? bnz : bnx;   bN.y = half ? 0.0f : bny;

    const v8f zero = {};
    // G1 = P.P^T, G2 = D.P^T, G3 = Nrm.Nrm^T  (all fp32, K padded 3->4)
    v8f g1 = __builtin_amdgcn_wmma_f32_16x16x4_f32(false, aP, false, bP, (short)0, zero, false, false);
    v8f g2 = __builtin_amdgcn_wmma_f32_16x16x4_f32(false, aD, false, bP, (short)0, zero, false, false);
    v8f g3 = __builtin_amdgcn_wmma_f32_16x16x4_f32(false, aN, false, bN, (short)0, zero, false, false);

    // ---- per-column (j) scalars, reusing the bP loads ----
    const float ppj   = bpx * bpx + bpy * bpy + bpz * bpz;
    const float densj = Rb[j];

    const float a1 = *a1p, a2 = *a2p, a3 = *a3p;

    // ---- epilogue over the 8 C/D VGPRs: row i = i0 + 8*half + r, col j ----
    const int iBase = i0 + (half << 3);
    float* __restrict__ outRow = out + ((size_t)b * N + iBase) * N + j;
    const float* __restrict__ Prow = Pb + (size_t)iBase * 3;
    const float* __restrict__ Drow = Db + (size_t)iBase * 3;
    const float* __restrict__ Rrow = Rb + iBase;

#pragma unroll
    for (int r = 0; r < 8; ++r) {
        const float pix = Prow[r * 3 + 0];
        const float piy = Prow[r * 3 + 1];
        const float piz = Prow[r * 3 + 2];
        const float dix = Drow[r * 3 + 0];
        const float diy = Drow[r * 3 + 1];
        const float diz = Drow[r * 3 + 2];
        const float ppi   = pix * pix + piy * piy + piz * piz;
        const float pdotd = pix * dix + piy * diy + piz * diz;
        const float densi = Rrow[r];

        const float along = pdotd - g2[r];
        const float dist2 = ppi + ppj - 2.0f * g1[r];
        const float perp2 = dist2 - along * along;

        // clip(sqrt(x), lo, hi) == sqrt(clip(x, lo^2, hi^2)); clamp also
        // guards the negative-cancellation case, and keeps the sqrt input in
        // a safe normal range so the raw v_sqrt_f32 is sufficient.
        const float p2c = fminf(fmaxf(perp2, W_MIN * W_MIN), W_MAX * W_MAX);
        const float nd  = __builtin_amdgcn_sqrtf(p2c);

        const float gate = a1 * (densi - densj) + a2 * nd - a3 * (1.0f - g3[r]);

        // sigmoid(gate) = 1 / (1 + exp2(-gate*log2(e)))
        const float e = __builtin_amdgcn_exp2f(gate * (-LOG2E));
        const float val = __builtin_amdgcn_rcpf(1.0f + e);

        outRow[(size_t)r * N] = val;
    }
}

extern "C" void kernel_launch(void* const* d_in, const int* in_sizes, int n_in,
                              void* d_out, int out_size, void* d_ws, size_t ws_size,
                              hipStream_t stream) {
    (void)n_in; (void)out_size; (void)d_ws; (void)ws_size;
    const float* points  = (const float*)d_in[0];
    const float* pdir    = (const float*)d_in[1];
    const float* normals = (const float*)d_in[2];
    const float* density = (const float*)d_in[3];
    const float* a1p     = (const float*)d_in[4];
    const float* a2p     = (const float*)d_in[5];
    const float* a3p     = (const float*)d_in[6];
    float* out = (float*)d_out;

    const int N = 2048;
    const int B = in_sizes[3] / N;          // density is [B,N]
    const int tilesPerSide  = N / 16;       // 128
    const int totalTiles    = B * tilesPerSide * tilesPerSide;
    const int wavesPerBlock = 8;            // 256 threads
    const int blocks = (totalTiles + wavesPerBlock - 1) / wavesPerBlock;

    SideGating_wmma_kernel<<<blocks, 256, 0, stream>>>(
        points, pdir, normals, density, a1p, a2p, a3p, out, N, tilesPerSide);
}